// SVRasterModel_11587821765186
// MI455X (gfx1250) — compile-verified
//
#include <hip/hip_runtime.h>
#include <hip/hip_bf16.h>
#include <stdint.h>

#define RES      128
#define NSAMP    128
#define NEAR_T   0.05f
#define FAR_T    4.0f
#define NUM_RAYS 32768
#define WPB      4                 // waves (rays) per block
#define BLOCK_T  (WPB * 32)

// CDNA5 async global->LDS copies (ASYNCcnt-tracked). `offset:` is added to
// BOTH the LDS dest address and the global address (CDNA5 ISA),
// so one LDS base + one 32-bit global offset covers a whole record.
#define ASY_B128(lds, voff, base, IMM)                                        \
  asm volatile("global_load_async_to_lds_b128 %0, %1, %2 offset:" IMM         \
               :: "v"(lds), "v"(voff), "s"(base) : "memory")
#define ASY_B64(lds, voff, base, IMM)                                         \
  asm volatile("global_load_async_to_lds_b64 %0, %1, %2 offset:" IMM          \
               :: "v"(lds), "v"(voff), "s"(base) : "memory")
#define ASY_B32(lds, voff, base, IMM)                                         \
  asm volatile("global_load_async_to_lds_b32 %0, %1, %2 offset:" IMM          \
               :: "v"(lds), "v"(voff), "s"(base) : "memory")

__global__ __launch_bounds__(BLOCK_T)
void svraster_fused(const float* __restrict__ rays_o,
                    const float* __restrict__ rays_d,
                    const float* __restrict__ vdens,
                    const float* __restrict__ vsh,
                    float* __restrict__ out)
{
    // Per-thread double-buffered staging slot: 27 SH floats + 1 density = 112B.
    __shared__ float smem[2][BLOCK_T][28];

    const int tid  = threadIdx.x;
    const int lane = tid & 31;
    const int wave = tid >> 5;
    const int ray  = blockIdx.x * WPB + wave;

    const float ox = rays_o[ray * 3 + 0];
    const float oy = rays_o[ray * 3 + 1];
    const float oz = rays_o[ray * 3 + 2];
    const float dx = rays_d[ray * 3 + 0];
    const float dy = rays_d[ray * 3 + 1];
    const float dz = rays_d[ray * 3 + 2];

    // SH basis up to degree 2 (wave-uniform: one ray per wave)
    float b[9];
    b[0] = 0.28209479177387814f;
    b[1] = -0.4886025119029199f * dy;
    b[2] =  0.4886025119029199f * dz;
    b[3] = -0.4886025119029199f * dx;
    b[4] =  1.0925484305920792f * dx * dy;
    b[5] = -1.0925484305920792f * dy * dz;
    b[6] =  0.31539156525252005f * (3.0f * dz * dz - 1.0f);
    b[7] = -1.0925484305920792f * dx * dz;
    b[8] =  0.5462742152960396f * (dx * dx - dy * dy);

    const float dt = (FAR_T - NEAR_T) / (float)(NSAMP - 1);

    uint32_t lds_base[2];
    lds_base[0] = (uint32_t)(uintptr_t)&smem[0][tid][0];  // low 32b = LDS offset
    lds_base[1] = (uint32_t)(uintptr_t)&smem[1][tid][0];

    // Sample j of this lane = global sample lane*4 + j (contiguous chunks so
    // the cross-lane scan preserves depth order).
    auto compute_idx = [&](int j, bool& in, uint32_t& voff_sh, uint32_t& voff_d) {
        float t  = NEAR_T + dt * (float)(lane * 4 + j);
        float px = fmaf(dx, t, ox);
        float py = fmaf(dy, t, oy);
        float pz = fmaf(dz, t, oz);
        float ux = (px + 1.0f) * 64.0f;   // (p - min)/(max-min) * RES
        float uy = (py + 1.0f) * 64.0f;
        float uz = (pz + 1.0f) * 64.0f;
        int ix = (int)floorf(ux);
        int iy = (int)floorf(uy);
        int iz = (int)floorf(uz);
        in = (ix >= 0) & (ix < RES) & (iy >= 0) & (iy < RES) & (iz >= 0) & (iz < RES);
        int cx = min(max(ix, 0), RES - 1);
        int cy = min(max(iy, 0), RES - 1);
        int cz = min(max(iz, 0), RES - 1);
        uint32_t flat = (uint32_t)((cx * RES + cy) * RES + cz);
        voff_sh = flat * 108u;   // 27 floats
        voff_d  = flat * 4u;
    };

    // 9 async instructions per stage (fixed count -> s_wait_asynccnt bookkeeping)
    auto issue = [&](int buf, uint32_t voff_sh, uint32_t voff_d) {
        uint32_t l = lds_base[buf];
        ASY_B128(l, voff_sh, vsh, "0");    // floats 0..3
        ASY_B128(l, voff_sh, vsh, "16");   // floats 4..7
        ASY_B128(l, voff_sh, vsh, "32");   // floats 8..11
        ASY_B128(l, voff_sh, vsh, "48");   // floats 12..15
        ASY_B128(l, voff_sh, vsh, "64");   // floats 16..19
        ASY_B128(l, voff_sh, vsh, "80");   // floats 20..23
        ASY_B64 (l, voff_sh, vsh, "96");   // floats 24..25
        ASY_B32 (l, voff_sh, vsh, "104");  // float  26
        uint32_t ld = l + 108u;
        ASY_B32 (ld, voff_d, vdens, "0");  // density -> slot float 27
    };

    bool  inb[4];
    float alpha[4], r0[4], r1[4], r2[4];

    // Prologue: stage 0 in flight.
    {
        uint32_t vs, vd; bool in0;
        compute_idx(0, in0, vs, vd);
        inb[0] = in0;
        issue(0, vs, vd);
    }

#pragma unroll
    for (int j = 0; j < 4; ++j) {
        if (j < 3) {
            uint32_t vs, vd; bool ii;
            compute_idx(j + 1, ii, vs, vd);
            inb[j + 1] = ii;
            // WAR guard: async LDS writes are unordered vs indexed DS reads
            // (ISA 08 §4.1). Ensure the previous iteration's ds_loads from the
            // buffer we are about to overwrite have fully retired before the
            // new async writes are even issued.
            asm volatile("s_wait_dscnt 0x0" ::: "memory");
            issue((j + 1) & 1, vs, vd);
            // <=9 outstanding left == stage j+1's loads; stage j fully landed
            asm volatile("s_wait_asynccnt 0x9" ::: "memory");
        } else {
            asm volatile("s_wait_asynccnt 0x0" ::: "memory");
        }

        const float4* sp = (const float4*)&smem[j & 1][tid][0]; // 16B-aligned slot
        float sv[28];
        float4* svv = (float4*)sv;
#pragma unroll
        for (int k = 0; k < 7; ++k) svv[k] = sp[k];             // ds_load_b128 x7

        if (inb[j]) {
            float dens = sv[27];
            alpha[j] = 1.0f - __expf(-dt * __expf(dens));
#pragma unroll
            for (int c = 0; c < 3; ++c) {
                float acc = 0.0f;
#pragma unroll
                for (int k = 0; k < 9; ++k)
                    acc = fmaf(b[k], sv[k * 3 + c], acc);       // sh[k][c]
                float sg = __builtin_amdgcn_rcpf(1.0f + __expf(-acc));
                if      (c == 0) r0[j] = sg;
                else if (c == 1) r1[j] = sg;
                else             r2[j] = sg;
            }
        } else {
            alpha[j] = 0.0f; r0[j] = 0.0f; r1[j] = 0.0f; r2[j] = 0.0f;
        }
    }

    // ---- Transmittance: wave32 multiplicative prefix scan over lane products ----
    float P = 1.0f;
#pragma unroll
    for (int j = 0; j < 4; ++j) P *= (1.0f - alpha[j] + 1e-10f);

    float incl = P;
#pragma unroll
    for (int off = 1; off < 32; off <<= 1) {
        float up = __shfl_up(incl, off, 32);
        if (lane >= off) incl *= up;
    }
    float trans = __shfl_up(incl, 1, 32);   // exclusive prefix
    if (lane == 0) trans = 1.0f;

    // ---- Composite this lane's 4 samples ----
    float a0 = 0.0f, a1 = 0.0f, a2 = 0.0f;
#pragma unroll
    for (int j = 0; j < 4; ++j) {
        float w = alpha[j] * trans;
        a0 = fmaf(w, r0[j], a0);
        a1 = fmaf(w, r1[j], a1);
        a2 = fmaf(w, r2[j], a2);
        trans *= (1.0f - alpha[j] + 1e-10f);
    }

    // ---- Wave reduction of the 3 channels ----
#pragma unroll
    for (int off = 16; off >= 1; off >>= 1) {
        a0 += __shfl_down(a0, off, 32);
        a1 += __shfl_down(a1, off, 32);
        a2 += __shfl_down(a2, off, 32);
    }
    if (lane == 0) {
        out[ray * 3 + 0] = a0;
        out[ray * 3 + 1] = a1;
        out[ray * 3 + 2] = a2;
    }
}

extern "C" void kernel_launch(void* const* d_in, const int* in_sizes, int n_in,
                              void* d_out, int out_size, void* d_ws, size_t ws_size,
                              hipStream_t stream) {
    (void)in_sizes; (void)n_in; (void)out_size; (void)d_ws; (void)ws_size;
    const float* rays_o = (const float*)d_in[0];
    const float* rays_d = (const float*)d_in[1];
    const float* vdens  = (const float*)d_in[2];
    const float* vsh    = (const float*)d_in[3];
    float*       out    = (float*)d_out;

    dim3 grid(NUM_RAYS / WPB);   // 8192 blocks x 128 threads = 32768 waves
    dim3 block(BLOCK_T);
    svraster_fused<<<grid, block, 0, stream>>>(rays_o, rays_d, vdens, vsh, out);
}